// NeRFLoss_18880676233822
// MI455X (gfx1250) — compile-verified
//
#include <hip/hip_runtime.h>
#include <hip/hip_bf16.h>

// ---------------------------------------------------------------------------
// NeRF loss, MI455X (gfx1250, wave32).
// Memory-bound: ~105 MB traffic -> ~4.5 us at 23.3 TB/s HBM. One wave per ray,
// float4 (b128) coalesced loads, DPP16-based wave32 scan (no LDS/bpermute),
// WMMA-assisted final reduction.
// ---------------------------------------------------------------------------

#define LAMBDA_OPACITY    1e-3f
#define LAMBDA_DISTORTION 1e-3f

typedef __attribute__((ext_vector_type(2))) float v2f;
typedef __attribute__((ext_vector_type(8))) float v8f;

// v_add_f32 with DPP row_shr:<N>, bound_ctrl:1 (invalid lanes contribute 0).
// CTRL: 0x110 | n  == row_shr:n  (rows of 16 lanes; wave32 has 2 rows).
template <int CTRL>
__device__ __forceinline__ float dpp_add(float x) {
  int s = __builtin_amdgcn_update_dpp(0, __builtin_bit_cast(int, x),
                                      CTRL, 0xf, 0xf, /*bound_ctrl=*/true);
  return x + __builtin_bit_cast(float, s);
}

__device__ __forceinline__ float bcast_lane(float x, int lane) {
  return __builtin_bit_cast(float,
      __builtin_amdgcn_readlane(__builtin_bit_cast(int, x), lane));
}

// --- Kernel 1: rgb squared error + opacity entropy (elementwise) -----------
__global__ void nerf_ew_kernel(const float* __restrict__ rgb,
                               const float* __restrict__ tgt,
                               const float* __restrict__ opac,
                               float* __restrict__ out_rgb,
                               float* __restrict__ out_op,
                               int n_rgb, int n_rays) {
  int tid = blockIdx.x * blockDim.x + threadIdx.x;
  if (tid < n_rgb) {
    float d = rgb[tid] - tgt[tid];
    out_rgb[tid] = d * d;
  } else {
    int r = tid - n_rgb;
    if (r < n_rays) {
      float o = opac[r] + 1e-10f;
      out_op[r] = LAMBDA_OPACITY * (-o * logf(o));
    }
  }
}

// --- Kernel 2: per-ray distortion loss -------------------------------------
// loss_r = sum_i 2*w_i*(t_i*W_{<i} - WT_{<i}) + (1/3) sum_i w_i^2 * delta_i
// One wave per ray; lane holds 4 consecutive samples (128 = 32 lanes * 4).
__global__ void nerf_dist_kernel(const float* __restrict__ w_g,
                                 const float* __restrict__ delta_g,
                                 const float* __restrict__ t_g,
                                 const int*   __restrict__ rays_a,
                                 float* __restrict__ out,
                                 int n_rays) {
  const int lane = threadIdx.x & 31;
  const int ray  = blockIdx.x * (blockDim.x >> 5) + (threadIdx.x >> 5);
  if (ray >= n_rays) return;          // uniform per wave: EXEC stays all-ones

  const int ray_id = rays_a[3 * ray + 0];
  const int start  = rays_a[3 * ray + 1];
  const int ns     = rays_a[3 * ray + 2];   // 128 in this problem

  float carry_w = 0.0f, carry_v = 0.0f, acc = 0.0f;

  for (int base = 0; base < ns; base += 128) {
    const int j = start + base + lane * 4;
    const float4 w4 = *reinterpret_cast<const float4*>(w_g + j);
    const float4 t4 = *reinterpret_cast<const float4*>(t_g + j);
    const float4 d4 = *reinterpret_cast<const float4*>(delta_g + j);

    const float v0 = w4.x * t4.x, v1 = w4.y * t4.y;
    const float v2 = w4.z * t4.z, v3 = w4.w * t4.w;
    const float lw = w4.x + w4.y + w4.z + w4.w;   // lane totals
    const float lv = v0 + v1 + v2 + v3;

    // Inclusive wave32 scan of lane totals: 4 DPP row_shr adds per value
    // (rows of 16), then add row-0 total into the upper row via readlane.
    float sw = lw, sv = lv;
    sw = dpp_add<0x111>(sw); sv = dpp_add<0x111>(sv);  // row_shr:1
    sw = dpp_add<0x112>(sw); sv = dpp_add<0x112>(sv);  // row_shr:2
    sw = dpp_add<0x114>(sw); sv = dpp_add<0x114>(sv);  // row_shr:4
    sw = dpp_add<0x118>(sw); sv = dpp_add<0x118>(sv);  // row_shr:8
    const float row0_w = bcast_lane(sw, 15);
    const float row0_v = bcast_lane(sv, 15);
    sw += (lane >= 16) ? row0_w : 0.0f;
    sv += (lane >= 16) ? row0_v : 0.0f;

    float ew = carry_w + (sw - lw);   // exclusive prefix entering this lane
    float ev = carry_v + (sv - lv);

    // Serial within-lane exclusive combine over the 4 samples.
    acc += 2.0f * w4.x * (t4.x * ew - ev) + w4.x * w4.x * d4.x * (1.0f / 3.0f);
    ew += w4.x; ev += v0;
    acc += 2.0f * w4.y * (t4.y * ew - ev) + w4.y * w4.y * d4.y * (1.0f / 3.0f);
    ew += w4.y; ev += v1;
    acc += 2.0f * w4.z * (t4.z * ew - ev) + w4.z * w4.z * d4.z * (1.0f / 3.0f);
    ew += w4.z; ev += v2;
    acc += 2.0f * w4.w * (t4.w * ew - ev) + w4.w * w4.w * d4.w * (1.0f / 3.0f);

    carry_w += bcast_lane(sw, 31);    // chunk totals -> next-chunk carry
    carry_v += bcast_lane(sv, 31);
  }

  // Wave reduction via one f32 WMMA: with A == all-ones(16x4), every row of
  // D equals the column sums of B (true for ANY bijective B operand layout).
  // D vgpr0, lanes 0..15 hold colsum_0..15; summing those 16 values yields
  // the sum of ALL 64 B slots = sum_lanes(acc) (the second B slot is 0).
  v2f a; a[0] = 1.0f; a[1] = 1.0f;
  v2f b; b[0] = acc;  b[1] = 0.0f;
  v8f c = {};
  c = __builtin_amdgcn_wmma_f32_16x16x4_f32(
      /*neg_a=*/false, a, /*neg_b=*/false, b,
      /*c_mod=*/(short)0, c, /*reuse_a=*/false, /*reuse_b=*/false);
  float r = c[0];
  r = dpp_add<0x111>(r);
  r = dpp_add<0x112>(r);
  r = dpp_add<0x114>(r);
  r = dpp_add<0x118>(r);                  // lane 15: sum of colsums 0..15
  const float total = bcast_lane(r, 15);  // == sum_lanes(acc)

  if (lane == 0) out[ray_id] = LAMBDA_DISTORTION * total;
}

// ---------------------------------------------------------------------------
extern "C" void kernel_launch(void* const* d_in, const int* in_sizes, int n_in,
                              void* d_out, int out_size, void* d_ws, size_t ws_size,
                              hipStream_t stream) {
  const float* rgb   = (const float*)d_in[0];
  const float* tgt   = (const float*)d_in[1];
  const float* opac  = (const float*)d_in[2];
  const float* w_g   = (const float*)d_in[3];
  const float* del_g = (const float*)d_in[4];
  const float* t_g   = (const float*)d_in[5];
  const int*   rays  = (const int*)d_in[6];

  const int n_rgb  = in_sizes[0];   // 196608
  const int n_rays = in_sizes[2];   // 65536

  float* out      = (float*)d_out;
  float* out_rgb  = out;
  float* out_op   = out + n_rgb;
  float* out_dist = out + n_rgb + n_rays;

  const int blk = 256;  // 8 wave32 waves per block

  const int total_ew = n_rgb + n_rays;
  nerf_ew_kernel<<<(total_ew + blk - 1) / blk, blk, 0, stream>>>(
      rgb, tgt, opac, out_rgb, out_op, n_rgb, n_rays);

  const int waves_per_block = blk / 32;
  const int blocks = (n_rays + waves_per_block - 1) / waves_per_block;
  nerf_dist_kernel<<<blocks, blk, 0, stream>>>(
      w_g, del_g, t_g, rays, out_dist, n_rays);
}